// AGAOperator_55533927137805
// MI455X (gfx1250) — compile-verified
//
#include <hip/hip_runtime.h>
#include <math.h>

typedef __attribute__((ext_vector_type(2))) float v2f;
typedef __attribute__((ext_vector_type(8))) float v8f;

#define NB 4
#define SS 2048
#define HH 1024
#define BOT 128
#define RD 48
#define NSLOT 4096
#define KK 8
#define FF 256               // H/4
#define NTOK (NB*SS)         // 8192
#define TILE 16
#define HPAD 1028            // 1024 + 4 pad (bank-conflict-free WMMA A reads)
#define QPAD 132             // 128 + 4
#define RPAD 49              // 48 + 1 (odd stride -> distinct banks)
#define RSQRT_RD 0.14433756729740643f

__device__ __forceinline__ v8f wmma4(v2f a, v2f b, v8f c) {
    return __builtin_amdgcn_wmma_f32_16x16x4_f32(false, a, false, b, (short)0, c, false, false);
}

__device__ __forceinline__ float sigmoidf_(float x) { return 1.0f / (1.0f + __expf(-x)); }

__device__ __forceinline__ void top8_insert(float (&v)[8], int (&ix)[8], float nv, int ni) {
    if (nv <= v[7]) return;
    int p = 7;
    while (p > 0 && v[p - 1] < nv) { v[p] = v[p - 1]; ix[p] = ix[p - 1]; --p; }
    v[p] = nv; ix[p] = ni;
}

// ---------------- Kernel 1: rk[n][r] = aux_keys[n][:] . router_w[r][:] ----------------
__global__ void __launch_bounds__(256) rk_kernel(const float* __restrict__ aux_keys,
                                                 const float* __restrict__ router_w,
                                                 float* __restrict__ rk) {
    int idx = blockIdx.x * 256 + threadIdx.x;
    if (idx >= NSLOT * RD) return;
    int n = idx / RD, r = idx % RD;
    const float* ak = aux_keys + (size_t)n * BOT;
    const float* wr = router_w + (size_t)r * BOT;
    float acc = 0.f;
#pragma unroll 8
    for (int i = 0; i < BOT; ++i) acc = fmaf(ak[i], wr[i], acc);
    rk[n * RD + r] = acc;
}

// ---------------- Kernel 2: fused per-16-token tile (WMMA heavy, pipelined) ----------------
__global__ void __launch_bounds__(256) token_kernel(
    const float* __restrict__ hs, const float* __restrict__ Wq,
    const float* __restrict__ Wr, const float* __restrict__ rk,
    const float* __restrict__ rel,
    const float* __restrict__ Wu1, const float* __restrict__ bu1,
    const float* __restrict__ Wu2, const float* __restrict__ bu2,
    float* __restrict__ topw, int* __restrict__ topidx,
    float* __restrict__ learned_out, float* __restrict__ logvar_out) {

    extern __shared__ float smem[];
    float* hid    = smem;                          // TILE*HPAD
    float* qld    = hid + TILE * HPAD;             // TILE*QPAD
    float* rqld   = qld + TILE * QPAD;             // TILE*RPAD
    float* rell   = rqld + TILE * RPAD;            // NSLOT
    float* cand_v = rell + NSLOT;                  // 8*32*8
    int*   cand_i = (int*)(cand_v + 8 * 32 * 8);   // 8*32*8
    float* lrn    = (float*)(cand_i + 8 * 32 * 8); // 16

    const int tid  = threadIdx.x;
    const int wave = tid >> 5;
    const int lane = tid & 31;
    const int mrow = lane & 15;           // M (or N) index within 16
    const int khalf = (lane >> 4) * 2;    // 0 for lanes 0-15, 2 for lanes 16-31
    const int trow = (lane >> 4) * 8;     // C-matrix M base for this lane half
    const int tok0 = blockIdx.x * TILE;

    // --- stage reliability + hidden tile into LDS ---
    for (int i = tid; i < NSLOT; i += 256) rell[i] = rel[i];
    for (int i = tid; i < TILE * (HH / 4); i += 256) {
        int t = i >> 8;          // / 256
        int h4 = i & 255;
        float4 v = ((const float4*)(hs + (size_t)(tok0 + t) * HH))[h4];
        float* dst = hid + t * HPAD + h4 * 4;
        dst[0] = v.x; dst[1] = v.y; dst[2] = v.z; dst[3] = v.w;
    }
    if (tid < 16) lrn[tid] = 0.f;
    __syncthreads();

    const float* ap0 = hid + mrow * HPAD + khalf;  // A stream (LDS) used by Q and UNC gemms

    // --- Q = hidden_tile (16xH) * Wq^T (Hx128); wave w owns output cols [16w,16w+16) ---
    {
        v8f acc = {};
        const float* wp0 = Wq + (size_t)(wave * 16 + mrow) * HH + khalf;
        v2f a0[8], b0[8], a1[8], b1[8];
        auto ldA = [&](v2f (&a)[8], int kb) {
#pragma unroll
            for (int j = 0; j < 8; ++j) { const float* p = ap0 + kb + j * 4; a[j].x = p[0]; a[j].y = p[1]; }
        };
        auto ldB = [&](v2f (&b)[8], int kb) {
#pragma unroll
            for (int j = 0; j < 8; ++j) { const float* p = wp0 + kb + j * 4; b[j].x = p[0]; b[j].y = p[1]; }
        };
        ldA(a0, 0); ldB(b0, 0);
        for (int kb = 0; kb < HH; kb += 64) {
            ldA(a1, kb + 32); ldB(b1, kb + 32);
            __builtin_prefetch(wp0 + kb + 256, 0, 1);
#pragma unroll
            for (int j = 0; j < 8; ++j) acc = wmma4(a0[j], b0[j], acc);
            int kn = (kb + 64 < HH) ? kb + 64 : 0;   // tail: dead dummy load
            ldA(a0, kn); ldB(b0, kn);
#pragma unroll
            for (int j = 0; j < 8; ++j) acc = wmma4(a1[j], b1[j], acc);
        }
        int dcol = wave * 16 + mrow;
#pragma unroll
        for (int r = 0; r < 8; ++r) qld[(trow + r) * QPAD + dcol] = acc[r];
    }
    __syncthreads();

    // --- rq = Q (16x128) * Wr^T (128x48), scaled by 1/sqrt(RD); waves 0..2 ---
    if (wave < 3) {
        v8f acc = {};
        const float* qp0 = qld + mrow * QPAD + khalf;
        const float* wp0 = Wr + (size_t)(wave * 16 + mrow) * BOT + khalf;
        for (int kb = 0; kb < BOT; kb += 32) {
            v2f a[8], b[8];
#pragma unroll
            for (int j = 0; j < 8; ++j) {
                const float* p = qp0 + kb + j * 4; a[j].x = p[0]; a[j].y = p[1];
                const float* q = wp0 + kb + j * 4; b[j].x = q[0]; b[j].y = q[1];
            }
#pragma unroll
            for (int j = 0; j < 8; ++j) acc = wmma4(a[j], b[j], acc);
        }
        int rcol = wave * 16 + mrow;
#pragma unroll
        for (int r = 0; r < 8; ++r) rqld[(trow + r) * RPAD + rcol] = acc[r] * RSQRT_RD;
    }
    __syncthreads();

    // --- Scores transposed: S^T = rk (NSLOTx48) x rq^T (48x16tok); fused top-8 ---
    // B (rq^T) is loop-invariant: hoist 12 ksteps into registers.
    v2f bmat[12];
#pragma unroll
    for (int ks = 0; ks < 12; ++ks) {
        int k = ks * 4 + khalf;
        bmat[ks].x = rqld[mrow * RPAD + k];
        bmat[ks].y = rqld[mrow * RPAD + k + 1];
    }
    float bv[8]; int bi[8];
#pragma unroll
    for (int j = 0; j < 8; ++j) { bv[j] = -3.4e38f; bi[j] = 0; }

    {
        auto ldRK = [&](v2f (&a)[12], int nt_) {
            const float* rkp = rk + (size_t)(nt_ * 16 + mrow) * RD + khalf;
#pragma unroll
            for (int ks = 0; ks < 12; ++ks) { a[ks].x = rkp[ks * 4]; a[ks].y = rkp[ks * 4 + 1]; }
        };
        auto procTile = [&](v2f (&a)[12], int nt_) {
            int srow = nt_ * 16 + trow;
            v8f acc;
#pragma unroll
            for (int r = 0; r < 8; ++r) acc[r] = rell[srow + r];   // fold reliability into C
#pragma unroll
            for (int ks = 0; ks < 12; ++ks) acc = wmma4(a[ks], bmat[ks], acc);
            // lane holds scores for token=mrow, slots srow..srow+7
#pragma unroll
            for (int r = 0; r < 8; ++r) top8_insert(bv, bi, acc[r], srow + r);
        };
        v2f aA[12], aB[12];
        int nt = wave;
        ldRK(aA, nt);
        for (int it = 0; it < 16; ++it) {         // 32 tiles per wave, 2 per iteration
            ldRK(aB, nt + 8);
            procTile(aA, nt);
            int ntN = (nt + 16 < NSLOT / 16) ? nt + 16 : wave;  // tail: dead dummy load
            ldRK(aA, ntN);
            procTile(aB, nt + 8);
            nt += 16;
        }
    }
#pragma unroll
    for (int j = 0; j < 8; ++j) {
        cand_v[(wave * 32 + lane) * 8 + j] = bv[j];
        cand_i[(wave * 32 + lane) * 8 + j] = bi[j];
    }
    __syncthreads();

    // --- final top-8 merge + softmax: thread t<16 owns token t (candidates live in lanes t, t+16 of every wave) ---
    if (tid < 16) {
        float fv[8]; int fi[8];
#pragma unroll
        for (int j = 0; j < 8; ++j) { fv[j] = -3.4e38f; fi[j] = 0; }
        for (int w = 0; w < 8; ++w)
            for (int half = 0; half < 2; ++half) {
                int base = (w * 32 + tid + half * 16) * 8;
#pragma unroll
                for (int j = 0; j < 8; ++j) top8_insert(fv, fi, cand_v[base + j], cand_i[base + j]);
            }
        float m = fv[0], s = 0.f;
#pragma unroll
        for (int j = 0; j < 8; ++j) s += __expf(fv[j] - m);
        float inv = 1.0f / s;
        size_t o = (size_t)(tok0 + tid) * KK;
#pragma unroll
        for (int j = 0; j < 8; ++j) {
            topw[o + j] = __expf(fv[j] - m) * inv;
            topidx[o + j] = fi[j];
        }
    }

    // --- uncertainty MLP: h1 = gelu(hidden*Wu1^T + b1); learned += h1*Wu2 (fused) ---
    {
        v8f accL = {};
        for (int nt = wave; nt < FF / 16; nt += 8) {
            int fcol = nt * 16 + mrow;
            float bias = bu1[fcol];
            v8f acc;
#pragma unroll
            for (int r = 0; r < 8; ++r) acc[r] = bias;
            const float* wp0 = Wu1 + (size_t)fcol * HH + khalf;
            v2f a0[8], b0[8], a1[8], b1[8];
            auto ldA = [&](v2f (&a)[8], int kb) {
#pragma unroll
                for (int j = 0; j < 8; ++j) { const float* p = ap0 + kb + j * 4; a[j].x = p[0]; a[j].y = p[1]; }
            };
            auto ldB = [&](v2f (&b)[8], int kb) {
#pragma unroll
                for (int j = 0; j < 8; ++j) { const float* p = wp0 + kb + j * 4; b[j].x = p[0]; b[j].y = p[1]; }
            };
            ldA(a0, 0); ldB(b0, 0);
            for (int kb = 0; kb < HH; kb += 64) {
                ldA(a1, kb + 32); ldB(b1, kb + 32);
                __builtin_prefetch(wp0 + kb + 256, 0, 1);
#pragma unroll
                for (int j = 0; j < 8; ++j) acc = wmma4(a0[j], b0[j], acc);
                int kn = (kb + 64 < HH) ? kb + 64 : 0;   // tail: dead dummy load
                ldA(a0, kn); ldB(b0, kn);
#pragma unroll
                for (int j = 0; j < 8; ++j) acc = wmma4(a1[j], b1[j], acc);
            }
            float w2 = Wu2[fcol];
#pragma unroll
            for (int r = 0; r < 8; ++r) {
                float x = acc[r];
                float g = 0.5f * x * (1.0f + erff(x * 0.70710678118f));  // exact gelu
                accL[r] += g * w2;
            }
        }
#pragma unroll
        for (int r = 0; r < 8; ++r) atomicAdd(&lrn[trow + r], accL[r]);  // ds_add_f32
    }
    __syncthreads();
    if (tid < 16) learned_out[tok0 + tid] = lrn[tid] + bu2[0];

    // --- per-token mean/variance -> log1p(var); wave w owns tokens 2w, 2w+1 ---
    {
        int half = lane >> 4;
        int t = wave * 2 + half;
        int li = lane & 15;
        float s = 0.f, ss = 0.f;
        for (int i = li; i < HH; i += 16) {
            float x = hid[t * HPAD + i];
            s += x; ss = fmaf(x, x, ss);
        }
#pragma unroll
        for (int off = 1; off < 16; off <<= 1) {
            s += __shfl_xor(s, off, 32);
            ss += __shfl_xor(ss, off, 32);
        }
        if (li == 0) {
            float mean = s * (1.0f / HH);
            float var = ss * (1.0f / HH) - mean * mean;
            logvar_out[tok0 + t] = log1pf(var);
        }
    }
}

// ---------------- Kernel 3: global mean of log-variance ----------------
__global__ void __launch_bounds__(256) reduce_kernel(const float* __restrict__ lv,
                                                     float* __restrict__ out) {
    __shared__ float s[256];
    float a = 0.f;
    for (int i = threadIdx.x; i < NTOK; i += 256) a += lv[i];
    s[threadIdx.x] = a;
    __syncthreads();
    for (int st = 128; st > 0; st >>= 1) {
        if (threadIdx.x < st) s[threadIdx.x] += s[threadIdx.x + st];
        __syncthreads();
    }
    if (threadIdx.x == 0) out[0] = s[0] * (1.0f / NTOK);
}

// ---------------- Kernel 4: gate + gather-aggregate + residual ----------------
__global__ void __launch_bounds__(256) final_kernel(
    const float* __restrict__ hs, const float* __restrict__ aux_values,
    const float* __restrict__ topw, const int* __restrict__ topidx,
    const float* __restrict__ learned, const float* __restrict__ logvar,
    const float* __restrict__ lvmean, const float* __restrict__ gw1,
    const float* __restrict__ gb, float* __restrict__ out) {

    __shared__ float sw[8];
    __shared__ int si[8];
    __shared__ float sg;
    const int token = blockIdx.x;
    const int tid = threadIdx.x;
    if (tid < 8) {
        sw[tid] = topw[(size_t)token * KK + tid];
        si[tid] = topidx[(size_t)token * KK + tid];
    }
    if (tid == 0) {
        float nvar = logvar[token] / (lvmean[0] + 1e-6f);
        float unc = nvar * 0.5f + sigmoidf_(learned[token]) * 2.5f;
        unc = fminf(fmaxf(unc, 0.0f), 5.0f);
        sg = sigmoidf_(gw1[0] * unc + gb[0]);
    }
    __syncthreads();
    float gate = sg;
    const float4* h4 = (const float4*)(hs + (size_t)token * HH);
    float4* o4 = (float4*)(out + (size_t)token * HH);
    float4 h = h4[tid];
    float4 agg = make_float4(0.f, 0.f, 0.f, 0.f);
#pragma unroll
    for (int k = 0; k < KK; ++k) {
        const float4 v = ((const float4*)(aux_values + (size_t)si[k] * HH))[tid];
        float w = sw[k];
        agg.x = fmaf(w, v.x, agg.x);
        agg.y = fmaf(w, v.y, agg.y);
        agg.z = fmaf(w, v.z, agg.z);
        agg.w = fmaf(w, v.w, agg.w);
    }
    float4 r;
    r.x = h.x + gate * agg.x;
    r.y = h.y + gate * agg.y;
    r.z = h.z + gate * agg.z;
    r.w = h.w + gate * agg.w;
    o4[tid] = r;
}

extern "C" void kernel_launch(void* const* d_in, const int* in_sizes, int n_in,
                              void* d_out, int out_size, void* d_ws, size_t ws_size,
                              hipStream_t stream) {
    const float* hs  = (const float*)d_in[0];
    const float* Wq  = (const float*)d_in[1];
    const float* Wr  = (const float*)d_in[2];
    const float* AK  = (const float*)d_in[3];
    const float* AV  = (const float*)d_in[4];
    const float* rel = (const float*)d_in[5];
    const float* Wu1 = (const float*)d_in[6];
    const float* bu1 = (const float*)d_in[7];
    const float* Wu2 = (const float*)d_in[8];
    const float* bu2 = (const float*)d_in[9];
    const float* gw1 = (const float*)d_in[10];
    const float* gb  = (const float*)d_in[11];
    float* out = (float*)d_out;

    float* ws      = (float*)d_ws;
    float* rk      = ws;                          // NSLOT*RD
    float* topw    = rk + NSLOT * RD;             // NTOK*KK
    int*   topidx  = (int*)(topw + NTOK * KK);    // NTOK*KK
    float* learned = (float*)(topidx + NTOK * KK);// NTOK
    float* logvar  = learned + NTOK;              // NTOK
    float* lvmean  = logvar + NTOK;               // 1

    constexpr size_t SMEM = (size_t)(TILE * HPAD + TILE * QPAD + TILE * RPAD +
                                     NSLOT + 8 * 32 * 8 + 8 * 32 * 8 + 16) * 4;

    rk_kernel<<<(NSLOT * RD + 255) / 256, 256, 0, stream>>>(AK, Wr, rk);
    token_kernel<<<NTOK / TILE, 256, SMEM, stream>>>(hs, Wq, Wr, rk, rel, Wu1, bu1, Wu2, bu2,
                                                     topw, topidx, learned, logvar);
    reduce_kernel<<<1, 256, 0, stream>>>(logvar, lvmean);
    final_kernel<<<NTOK, 256, 0, stream>>>(hs, AV, topw, topidx, learned, logvar, lvmean,
                                           gw1, gb, out);
}